// Loss_calculater_70248485094000
// MI455X (gfx1250) — compile-verified
//
#include <hip/hip_runtime.h>
#include <math.h>

typedef float v2f __attribute__((ext_vector_type(2)));
typedef float v8f __attribute__((ext_vector_type(8)));

#define BB 16
#define GG 32
#define CC 80
#define POS_THR 0.5f
#define NEG_THR 0.4f
#define EPSF 1e-6f

// ---------------------------------------------------------------------------
// Exact f32 block reduction (256 threads) using the CDNA5 matrix pipe:
// chain 4x V_WMMA_F32_16X16X4_F32 with an all-ones B matrix so that
// D[m,n] = sum_k A[m,k]; any lane->A mapping is a valid reduction order.
// Returns the block total (valid in wave 0; thread 0 uses it).
// Deterministic: fixed order every invocation.
// ---------------------------------------------------------------------------
__device__ __forceinline__ float block_reduce_256(float v, float* sm) {
    sm[threadIdx.x] = v;
    __syncthreads();
    float total = 0.0f;
    if (threadIdx.x < 32) {              // wave 0, EXEC all ones inside
        const int lane = threadIdx.x;
        v8f acc = {};
        v2f ones; ones.x = 1.0f; ones.y = 1.0f;
#pragma unroll
        for (int i = 0; i < 4; ++i) {
            v2f a;
            a.x = sm[i * 64 + 2 * lane];
            a.y = sm[i * 64 + 2 * lane + 1];
            // 8 args: (neg_a, A, neg_b, B, c_mod, C, reuse_a, reuse_b)
            acc = __builtin_amdgcn_wmma_f32_16x16x4_f32(
                false, a, false, ones, (short)0, acc, false, false);
        }
        // lane L holds D[m = j + (L<16?0:8), n = L%16]; every n column equals
        // the row sum, so total = laneSum(lane0) + laneSum(lane16).
        float s = acc[0] + acc[1] + acc[2] + acc[3] +
                  acc[4] + acc[5] + acc[6] + acc[7];
        total = __shfl(s, 0, 32) + __shfl(s, 16, 32);
    }
    __syncthreads();                     // sm safe to reuse after return
    return total;
}

// ---------------------------------------------------------------------------
// Kernel 1: per-anchor max-IoU assignment, flags+label, smooth-L1 reg loss.
// ---------------------------------------------------------------------------
__global__ void assign_kernel(const float* __restrict__ reg_pred,   // [B,A,4]
                              const float* __restrict__ anchors,    // [A,4]
                              const float* __restrict__ gt_boxes,   // [B,G,4]
                              const int*   __restrict__ gt_labels,  // [B,G]
                              int*   __restrict__ flags,            // [B*A]
                              float* __restrict__ part_reg,
                              float* __restrict__ part_npos,
                              int A) {
    __shared__ float sm[256];
    const int idx = blockIdx.x * 256 + threadIdx.x;
    const int BA = BB * A;
    float reg_loss = 0.0f;
    float npos = 0.0f;

    if (idx < BA) {
        const int b = idx / A;
        const int a = idx - b * A;
        const float ax1 = anchors[a * 4 + 0];
        const float ay1 = anchors[a * 4 + 1];
        const float ax2 = anchors[a * 4 + 2];
        const float ay2 = anchors[a * 4 + 3];
        const float area_a = (ax2 - ax1) * (ay2 - ay1);
        const float* gb = gt_boxes + b * GG * 4;

        float best = -1.0f;
        int bg = 0;
        for (int g = 0; g < GG; ++g) {
            const float gx1 = gb[g * 4 + 0];
            const float gy1 = gb[g * 4 + 1];
            const float gx2 = gb[g * 4 + 2];
            const float gy2 = gb[g * 4 + 3];
            float w = fminf(ax2, gx2) - fmaxf(ax1, gx1);
            float h = fminf(ay2, gy2) - fmaxf(ay1, gy1);
            w = fmaxf(w, 0.0f);
            h = fmaxf(h, 0.0f);
            const float inter  = w * h;
            const float area_g = (gx2 - gx1) * (gy2 - gy1);
            const float iou = inter / (area_a + area_g - inter + EPSF);
            if (iou > best) { best = iou; bg = g; }   // first-max like argmax
        }

        const bool pos = best >= POS_THR;
        const bool neg = best <  NEG_THR;
        const int  lab = gt_labels[b * GG + bg];
        flags[idx] = (lab & 0xFF) | (pos ? 0x100 : 0) | (neg ? 0x200 : 0);

        if (pos) {
            const float mx1 = gb[bg * 4 + 0];
            const float my1 = gb[bg * 4 + 1];
            const float mx2 = gb[bg * 4 + 2];
            const float my2 = gb[bg * 4 + 3];
            const float wa = ax2 - ax1, ha = ay2 - ay1;
            const float xa = ax1 + 0.5f * wa, ya = ay1 + 0.5f * ha;
            const float wg = fmaxf(mx2 - mx1, EPSF);
            const float hg = fmaxf(my2 - my1, EPSF);
            const float xg = mx1 + 0.5f * wg, yg = my1 + 0.5f * hg;
            float t[4];
            t[0] = (xg - xa) / wa;
            t[1] = (yg - ya) / ha;
            t[2] = logf(wg / wa);
            t[3] = logf(hg / ha);
            const float* pp = reg_pred + (size_t)idx * 4;
#pragma unroll
            for (int j = 0; j < 4; ++j) {
                const float d = fabsf(pp[j] - t[j]);
                reg_loss += (d < 1.0f) ? 0.5f * d * d : d - 0.5f;  // BETA=1
            }
            npos = 1.0f;
        }
    }

    float tr = block_reduce_256(reg_loss, sm);
    if (threadIdx.x == 0) part_reg[blockIdx.x] = tr;
    float tn = block_reduce_256(npos, sm);
    if (threadIdx.x == 0) part_npos[blockIdx.x] = tn;
}

// ---------------------------------------------------------------------------
// Kernel 2: streaming focal loss over [B*A, C] logits, float4 coalesced.
// ---------------------------------------------------------------------------
__global__ void cls_kernel(const float4* __restrict__ cls_pred,  // [B*A*C/4]
                           const int*    __restrict__ flags,     // [B*A]
                           float* __restrict__ part_cls,
                           int n4) {                              // B*A*(C/4)
    __shared__ float sm[256];
    const int stride = gridDim.x * 256;
    float acc = 0.0f;
    for (int i = blockIdx.x * 256 + threadIdx.x; i < n4; i += stride) {
        const int ba = i / (CC / 4);
        const int f  = flags[ba];
        if (f & 0x300) {                         // weight = pos | neg
            const int  c0  = (i - ba * (CC / 4)) * 4;
            const int  lab = f & 0xFF;
            const bool pos = (f & 0x100) != 0;
            const float4 x4 = cls_pred[i];
            float xs[4] = {x4.x, x4.y, x4.z, x4.w};
#pragma unroll
            for (int j = 0; j < 4; ++j) {
                const float x   = xs[j];
                const bool  tgt = pos && ((c0 + j) == lab);
                const float e   = expf(-fabsf(x));
                // logaddexp(0,x) = max(x,0) + log1p(exp(-|x|))
                const float ce  = fmaxf(x, 0.0f) + log1pf(e) - (tgt ? x : 0.0f);
                const float p   = (x >= 0.0f) ? 1.0f / (1.0f + e)
                                              : e / (1.0f + e);
                const float pt  = tgt ? p : 1.0f - p;
                const float om  = 1.0f - pt;
                const float at  = tgt ? 0.25f : 0.75f;   // ALPHA
                acc += at * om * om * ce;                // GAMMA = 2
            }
        }
    }
    float t = block_reduce_256(acc, sm);
    if (threadIdx.x == 0) part_cls[blockIdx.x] = t;
}

// ---------------------------------------------------------------------------
// Kernel 3: single-block final reduction + normalization (fixed order).
// ---------------------------------------------------------------------------
__global__ void final_kernel(const float* __restrict__ part_reg,
                             const float* __restrict__ part_npos,
                             const float* __restrict__ part_cls,
                             int nb1, int nb2,
                             float* __restrict__ out) {
    __shared__ float sm[256];
    float r = 0.0f, n = 0.0f, c = 0.0f;
    for (int i = threadIdx.x; i < nb1; i += 256) {
        r += part_reg[i];
        n += part_npos[i];
    }
    for (int i = threadIdx.x; i < nb2; i += 256) c += part_cls[i];
    float tr = block_reduce_256(r, sm);
    float tn = block_reduce_256(n, sm);
    float tc = block_reduce_256(c, sm);
    if (threadIdx.x == 0) {
        const float np = fmaxf(tn, 1.0f);
        const float lr = tr / np;
        const float lc = tc / np;
        out[0] = lr + lc;
        out[1] = lr;
        out[2] = lc;
    }
}

// ---------------------------------------------------------------------------
extern "C" void kernel_launch(void* const* d_in, const int* in_sizes, int n_in,
                              void* d_out, int out_size, void* d_ws, size_t ws_size,
                              hipStream_t stream) {
    const float* proposals_reg = (const float*)d_in[0]; // [B,A,4]
    const float* proposals_cls = (const float*)d_in[1]; // [B,A,C]
    const float* anchors       = (const float*)d_in[2]; // [A,4]
    const float* gt_boxes      = (const float*)d_in[3]; // [B,G,4]
    const int*   gt_labels     = (const int*)  d_in[4]; // [B,G]

    const int A   = in_sizes[2] / 4;      // 25575
    const int BA  = BB * A;               // 409200
    const int nb1 = (BA + 255) / 256;     // blocks for assign
    const int nb2 = 4096;                 // blocks for cls streaming
    const int n4  = BA * (CC / 4);        // float4 count of logits

    // Workspace layout (all overwritten every call; no stale-state reliance):
    int*   flags     = (int*)d_ws;                               // BA ints
    float* part_reg  = (float*)((char*)d_ws + sizeof(int) * (size_t)BA);
    float* part_npos = part_reg + nb1;
    float* part_cls  = part_npos + nb1;

    assign_kernel<<<dim3(nb1), dim3(256), 0, stream>>>(
        proposals_reg, anchors, gt_boxes, gt_labels,
        flags, part_reg, part_npos, A);

    cls_kernel<<<dim3(nb2), dim3(256), 0, stream>>>(
        (const float4*)proposals_cls, flags, part_cls, n4);

    final_kernel<<<dim3(1), dim3(256), 0, stream>>>(
        part_reg, part_npos, part_cls, nb1, nb2, (float*)d_out);
}